// ExtrinsicCluster_22256520528830
// MI455X (gfx1250) — compile-verified
//
#include <hip/hip_runtime.h>
#include <math.h>

#define A_DIM   512
#define KL      4096          // K*L = 64*64
#define N_FR    8
#define TILE    1024          // elements per block
#define TPB     256           // 8 wave32s
#define EPT     4             // elements per thread

// ---------------------------------------------------------------------------
// Kernel 0: build frame tables in d_ws.
//   ws[0..71]  : R[8][9]  (row-major 3x3, R = Rz*Ry*Rx, frame 0 = identity)
//   ws[72..95] : T[8][3]  (frame 0 = zero)
// ---------------------------------------------------------------------------
__global__ void build_frames_kernel(const float* __restrict__ rot,
                                    const float* __restrict__ trans,
                                    float* __restrict__ ws) {
    int i = threadIdx.x;
    if (i >= N_FR) return;
    float psi = 0.f, th = 0.f, phi = 0.f, tx = 0.f, ty = 0.f, tz = 0.f;
    if (i > 0) {
        psi = rot[(i - 1) * 3 + 0];
        th  = rot[(i - 1) * 3 + 1];
        phi = rot[(i - 1) * 3 + 2];
        tx  = trans[(i - 1) * 3 + 0];
        ty  = trans[(i - 1) * 3 + 1];
        tz  = trans[(i - 1) * 3 + 2];
    }
    float s1 = sinf(psi), c1 = cosf(psi);
    float s2 = sinf(th),  c2 = cosf(th);
    float s3 = sinf(phi), c3 = cosf(phi);
    float* R = ws + i * 9;
    R[0] = c3 * c2; R[1] = c3 * s2 * s1 - s3 * c1; R[2] = c3 * s2 * c1 + s3 * s1;
    R[3] = s3 * c2; R[4] = s3 * s2 * s1 + c3 * c1; R[5] = s3 * s2 * c1 - c3 * s1;
    R[6] = -s2;     R[7] = c2 * s1;                R[8] = c2 * c1;
    float* T = ws + N_FR * 9 + i * 3;
    T[0] = tx; T[1] = ty; T[2] = tz;
}

__device__ __forceinline__ void cross3(float ax, float ay, float az,
                                       float bx, float by, float bz,
                                       float& cx, float& cy, float& cz) {
    cx = ay * bz - az * by;
    cy = az * bx - ax * bz;
    cz = ax * by - ay * bx;
}

// ---------------------------------------------------------------------------
// Kernel 1: per-element geometry. One block = one 1024-element tile of one 'a'.
// x0/x1 tiles are staged into LDS via the CDNA5 async-to-LDS engine.
// ---------------------------------------------------------------------------
__global__ __launch_bounds__(TPB)
void extrinsic_cluster_kernel(const float* __restrict__ x0g,
                              const float* __restrict__ x1g,
                              const int*   __restrict__ i0g,
                              const int*   __restrict__ i1g,
                              const float* __restrict__ ws,
                              float* __restrict__ out) {
    __shared__ float s_x0[TILE * 3];   // 12 KB
    __shared__ float s_x1[TILE * 3];   // 12 KB

    const int a  = blockIdx.y;
    const int e0 = blockIdx.x * TILE;
    const int t  = threadIdx.x;

    // ---- async DMA of this tile into LDS (ASYNCcnt path, gfx1250) ----------
    const float* g0 = x0g + ((size_t)a * KL + e0) * 3;   // 16B aligned
    const float* g1 = x1g + ((size_t)a * KL + e0) * 3;
    unsigned l0 = (unsigned)(size_t)(&s_x0[0]);          // low 32 bits = LDS byte offset
    unsigned l1 = (unsigned)(size_t)(&s_x1[0]);
#pragma unroll
    for (int k = 0; k < 3; ++k) {
        unsigned boff = (unsigned)(t + k * TPB) * 16u;   // 768 x b128 per array
        asm volatile("global_load_async_to_lds_b128 %0, %1, %2"
                     :: "v"(l0 + boff), "v"(boff), "s"(g0) : "memory");
        asm volatile("global_load_async_to_lds_b128 %0, %1, %2"
                     :: "v"(l1 + boff), "v"(boff), "s"(g1) : "memory");
    }

    // ---- per-'a' uniform parameters while the DMA is in flight -------------
    const int i0 = i0g[a];
    const int i1 = i1g[a];
    const float* Rt = ws;
    const float* Tt = ws + N_FR * 9;
    float R0[9], R1[9];
#pragma unroll
    for (int j = 0; j < 9; ++j) { R0[j] = Rt[i0 * 9 + j]; R1[j] = Rt[i1 * 9 + j]; }
    const float p0x = Tt[i0 * 3 + 0], p0y = Tt[i0 * 3 + 1], p0z = Tt[i0 * 3 + 2];
    const float p1x = Tt[i1 * 3 + 0], p1y = Tt[i1 * 3 + 1], p1z = Tt[i1 * 3 + 2];
    const float bx = p1x - p0x, by = p1y - p0y, bz = p1z - p0z;   // rhs

    // wait for this wave's async copies, then block-wide barrier
    asm volatile("s_wait_asynccnt 0x0" ::: "memory");
    __syncthreads();

    // ---- load 4 elements (12 floats = 3 x float4, 16B aligned) -------------
    const float4* r0 = (const float4*)(&s_x0[t * 12]);
    const float4* r1 = (const float4*)(&s_x1[t * 12]);
    float4 a0 = r0[0], a1 = r0[1], a2 = r0[2];
    float4 c0 = r1[0], c1v = r1[1], c2v = r1[2];
    float X0[EPT][3] = {{a0.x, a0.y, a0.z}, {a0.w, a1.x, a1.y},
                        {a1.z, a1.w, a2.x}, {a2.y, a2.z, a2.w}};
    float X1[EPT][3] = {{c0.x, c0.y, c0.z}, {c0.w, c1v.x, c1v.y},
                        {c1v.z, c1v.w, c2v.x}, {c2v.y, c2v.z, c2v.w}};

    float4 o0[EPT], o1[EPT];
#pragma unroll
    for (int e = 0; e < EPT; ++e) {
        const float x0x = X0[e][0], x0y = X0[e][1], x0z = X0[e][2];
        const float x1x = X1[e][0], x1y = X1[e][1], x1z = X1[e][2];
        // v = R @ x
        float v0x = R0[0] * x0x + R0[1] * x0y + R0[2] * x0z;
        float v0y = R0[3] * x0x + R0[4] * x0y + R0[5] * x0z;
        float v0z = R0[6] * x0x + R0[7] * x0y + R0[8] * x0z;
        float v1x = R1[0] * x1x + R1[1] * x1y + R1[2] * x1z;
        float v1y = R1[3] * x1x + R1[4] * x1y + R1[5] * x1z;
        float v1z = R1[6] * x1x + R1[7] * x1y + R1[8] * x1z;
        // c = v1 x v0 ; v2 = normalize(c)
        float cx, cy, cz;
        cross3(v1x, v1y, v1z, v0x, v0y, v0z, cx, cy, cz);
        float inl = rsqrtf(cx * cx + cy * cy + cz * cz);
        float v2x = cx * inl, v2y = cy * inl, v2z = cz * inl;
        // Cramer (rows r0=v0, r1=-v1, r2=v2):
        //   s = (b0*(r1xr2) + b1*(r2xr0) + b2*(r0xr1)) / det
        //   r0xr1 = v0 x (-v1) = v1 x v0 = c  (reuse)
        float ux, uy, uz, wx, wy, wz;
        cross3(-v1x, -v1y, -v1z, v2x, v2y, v2z, ux, uy, uz);  // r1 x r2
        cross3(v2x, v2y, v2z, v0x, v0y, v0z, wx, wy, wz);     // r2 x r0
        float det = v0x * ux + v0y * uy + v0z * uz;
        float inv = 1.0f / det;
        float t0 = (bx * ux + by * wx + bz * cx) * inv;
        float t1 = (bx * uy + by * wy + bz * cy) * inv;
        // positions
        float q0x = p0x + t0 * v0x, q0y = p0y + t0 * v0y, q0z = p0z + t0 * v0z;
        float q1x = p1x + t1 * v1x, q1y = p1y + t1 * v1y, q1z = p1z + t1 * v1z;
        // angles: d0 = normalize(b - t0*v0), d1 = normalize(-b - t1*v1)
        float d0x = bx - t0 * v0x, d0y = by - t0 * v0y, d0z = bz - t0 * v0z;
        float n0 = rsqrtf(d0x * d0x + d0y * d0y + d0z * d0z);
        float ang0 = (v1x * d0x + v1y * d0y + v1z * d0z) * n0;
        float d1x = -bx - t1 * v1x, d1y = -by - t1 * v1y, d1z = -bz - t1 * v1z;
        float n1 = rsqrtf(d1x * d1x + d1y * d1y + d1z * d1z);
        float ang1 = (v0x * d1x + v0y * d1y + v0z * d1z) * n1;
        o0[e] = make_float4(q0x, q0y, q0z, ang0);
        o1[e] = make_float4(q1x, q1y, q1z, ang1);
    }

    // ---- coalesced float4 stores: out[a, s, k, l, 4] ----------------------
    float4* p_out0 = (float4*)(out + ((size_t)(a * 2 + 0) * KL + e0 + t * EPT) * 4);
    float4* p_out1 = (float4*)(out + ((size_t)(a * 2 + 1) * KL + e0 + t * EPT) * 4);
#pragma unroll
    for (int e = 0; e < EPT; ++e) { p_out0[e] = o0[e]; p_out1[e] = o1[e]; }
}

// ---------------------------------------------------------------------------
extern "C" void kernel_launch(void* const* d_in, const int* in_sizes, int n_in,
                              void* d_out, int out_size, void* d_ws, size_t ws_size,
                              hipStream_t stream) {
    const float* x0    = (const float*)d_in[0];
    const float* x1    = (const float*)d_in[1];
    const float* rot   = (const float*)d_in[2];
    const float* trans = (const float*)d_in[3];
    const int*   i0    = (const int*)d_in[4];
    const int*   i1    = (const int*)d_in[5];
    float* ws  = (float*)d_ws;
    float* out = (float*)d_out;

    build_frames_kernel<<<1, 32, 0, stream>>>(rot, trans, ws);

    dim3 grid(KL / TILE, A_DIM);   // (4, 512)
    extrinsic_cluster_kernel<<<grid, TPB, 0, stream>>>(x0, x1, i0, i1, ws, out);
}